// TreeTraversalTreeImpl_50302656970963
// MI455X (gfx1250) — compile-verified
//
#include <hip/hip_runtime.h>
#include <stdint.h>

// ---- problem constants (from reference) ----
#define NUM_TREES  512
#define NUM_NODES  2047     // complete binary tree, depth 10
#define MAX_DEPTH  10
#define N_FEATURES 256
#define BATCH      8192

// ---- tiling ----
#define BLOCK   256                       // 8 waves (wave32)
#define SPB     128                       // samples per block tile
#define TPB     64                        // trees per block chunk
#define ROW_PAD 260                       // padded LDS row stride in floats (bank spread)
#define LDS_BYTES (SPB * ROW_PAD * 4)     // 133,120 B -> 2 WGs per 320KB WGP
#define PAIR_ITERS ((SPB * TPB) / BLOCK)  // 32 (sample,tree) pairs per thread
#define ILP 4                             // concurrent tree walks per thread

// Pack per-node data into one 16B record so the traversal does a single
// global_load_b128 gather per hop instead of 4 scalar gathers.
__global__ __launch_bounds__(BLOCK) void pack_nodes_kernel(
    const int*   __restrict__ feats,
    const float* __restrict__ ths,
    const int*   __restrict__ lefts,
    const int*   __restrict__ rights,
    int4* __restrict__ nodes, int n) {
  int i = blockIdx.x * BLOCK + threadIdx.x;
  if (i < n) {
    int4 v;
    v.x = feats[i];
    v.y = __float_as_int(ths[i]);
    v.z = lefts[i];          // tree-local child index (reference semantics)
    v.w = rights[i];
    nodes[i] = v;
  }
}

template <bool PACKED>
__global__ __launch_bounds__(BLOCK) void forest_kernel(
    const float* __restrict__ x,
    const int4*  __restrict__ nodes,
    const int*   __restrict__ feats,
    const float* __restrict__ ths,
    const int*   __restrict__ lefts,
    const int*   __restrict__ rights,
    const float* __restrict__ leafVals,
    float*       __restrict__ out) {
  extern __shared__ float xs[];          // [SPB][ROW_PAD]
  const int tid        = threadIdx.x;
  const int sampleBase = blockIdx.x * SPB;
  const int treeBase   = blockIdx.y * TPB;

  // ---- stage x tile (SPB rows x 256 floats) into padded LDS via CDNA5
  //      async global->LDS b128 copies (tracked by ASYNCcnt) ----
  {
    const float*   gsrc    = x + (size_t)sampleBase * N_FEATURES;
    const uint32_t ldsBase = (uint32_t)(uintptr_t)xs;  // low 32b of flat LDS ptr = LDS offset
    for (int c = tid; c < SPB * (N_FEATURES / 4); c += BLOCK) {
      int row = c >> 6;                 // 64 x 16B chunks per row
      int col = (c & 63) << 2;          // float index, multiple of 4 -> 16B aligned
      uint64_t ga = (uint64_t)(uintptr_t)(gsrc + row * N_FEATURES + col);
      uint32_t la = ldsBase + (uint32_t)((row * ROW_PAD + col) * 4);
      asm volatile("global_load_async_to_lds_b128 %0, %1, off"
                   :: "v"(la), "v"(ga) : "memory");
    }
    // warm this chunk's first trees in L2 while the async copy is in flight
    if (PACKED && (tid & 31) == 0)
      __builtin_prefetch(nodes + (size_t)(treeBase + (tid >> 5)) * NUM_NODES, 0, 0);
    asm volatile("s_wait_asynccnt 0" ::: "memory");
  }
  __syncthreads();

  // ---- traversal ----
  // Lanes = samples (one wave walks one tree region => gathers stay inside a
  // 32KB tree, top levels broadcast). Each thread walks ILP=4 trees at once:
  // four independent dependent-chains per thread to hide gather latency
  // (4 waves/SIMD x 4 chains = 16 outstanding gather chains per SIMD).
  for (int it = 0; it < PAIR_ITERS; it += ILP) {
    int pairId = it * BLOCK + tid;
    int sLocal = pairId & (SPB - 1);
    int tc0    = pairId >> 7;              // SPB == 128
    const float* __restrict__ xrow = xs + sLocal * ROW_PAD;

    int tg[ILP];
    int local[ILP];
#pragma unroll
    for (int k = 0; k < ILP; ++k) {
      tg[k]    = treeBase + tc0 + 2 * k;   // chains walk trees tc0, tc0+2, ...
      local[k] = 0;                        // root (nodes_offset == tg*NUM_NODES)
    }

    if (PACKED) {
      const int4* __restrict__ tree[ILP];
#pragma unroll
      for (int k = 0; k < ILP; ++k) tree[k] = nodes + (size_t)tg[k] * NUM_NODES;

      // hint the next iteration's tree chunk (one lane per wave)
      if ((tid & 31) == 0 && tc0 + 2 * ILP < TPB)
        __builtin_prefetch(nodes + (size_t)(treeBase + tc0 + 2 * ILP) * NUM_NODES, 0, 0);

#pragma unroll
      for (int d = 0; d < MAX_DEPTH; ++d) {
#pragma unroll
        for (int k = 0; k < ILP; ++k) {
          int4  n  = tree[k][local[k]];                 // global_load_b128 gather
          float fv = xrow[n.x];                         // ds_load_b32
          local[k] = (fv <= __int_as_float(n.y)) ? n.z : n.w;
        }
      }
    } else {
#pragma unroll
      for (int d = 0; d < MAX_DEPTH; ++d) {
#pragma unroll
        for (int k = 0; k < ILP; ++k) {
          int   g  = tg[k] * NUM_NODES + local[k];
          int   f  = feats[g];
          float th = ths[g];
          int   l  = lefts[g];
          int   r  = rights[g];
          float fv = xrow[f];
          local[k] = (fv <= th) ? l : r;
        }
      }
    }

#pragma unroll
    for (int k = 0; k < ILP; ++k) {
      out[(size_t)(sampleBase + sLocal) * NUM_TREES + tg[k]] =
          leafVals[(size_t)tg[k] * NUM_NODES + local[k]];
    }
  }
}

extern "C" void kernel_launch(void* const* d_in, const int* in_sizes, int n_in,
                              void* d_out, int out_size, void* d_ws, size_t ws_size,
                              hipStream_t stream) {
  (void)in_sizes; (void)n_in; (void)out_size;
  // setup_inputs() order: x, thresholds, values, lefts, rights, features,
  //                       nodes_offset, max_depth  (last two are static here)
  const float* x      = (const float*)d_in[0];
  const float* ths    = (const float*)d_in[1];
  const float* vals   = (const float*)d_in[2];
  const int*   lefts  = (const int*)d_in[3];
  const int*   rights = (const int*)d_in[4];
  const int*   feats  = (const int*)d_in[5];
  float* out = (float*)d_out;

  dim3 grid(BATCH / SPB, NUM_TREES / TPB);   // 64 x 8 = 512 workgroups
  const int nTot = NUM_TREES * NUM_NODES;    // 1,048,064 nodes

  if (ws_size >= (size_t)nTot * sizeof(int4)) {
    pack_nodes_kernel<<<(nTot + BLOCK - 1) / BLOCK, BLOCK, 0, stream>>>(
        feats, ths, lefts, rights, (int4*)d_ws, nTot);
    forest_kernel<true><<<grid, BLOCK, LDS_BYTES, stream>>>(
        x, (const int4*)d_ws, nullptr, nullptr, nullptr, nullptr, vals, out);
  } else {
    forest_kernel<false><<<grid, BLOCK, LDS_BYTES, stream>>>(
        x, nullptr, feats, ths, lefts, rights, vals, out);
  }
}